// FeynNet_59717225284402
// MI455X (gfx1250) — compile-verified
//
#include <hip/hip_runtime.h>

typedef __attribute__((ext_vector_type(2))) float v2f;
typedef __attribute__((ext_vector_type(8))) float v8f;

#define BDIM 128
#define CDIM 32
#define NA   12
#define NB   12
#define PDIM 8192
#define WAVES 8
#define TILES_PER_WAVE 4
#define EPS 1e-5f

// ---------------------------------------------------------------------------
// Kernel 0: fold BN1 + conv + BN2 into a single 32x32 matrix and 32-bias.
//   W'[o,c] = s2[o] * conv_w[o,c] * s1[c]
//   bias[o] = s2[o] * sum_c conv_w[o,c]*(beta1[c]-mean1[c]*s1[c]) + (beta2[o]-mean2[o]*s2[o])
// One wave32; thread o computes row o.
// ---------------------------------------------------------------------------
__global__ void feyn_fuse_params(const float* __restrict__ bn1_gamma,
                                 const float* __restrict__ bn1_beta,
                                 const float* __restrict__ bn1_mean,
                                 const float* __restrict__ bn1_var,
                                 const float* __restrict__ conv_w,
                                 const float* __restrict__ bn2_gamma,
                                 const float* __restrict__ bn2_beta,
                                 const float* __restrict__ bn2_mean,
                                 const float* __restrict__ bn2_var,
                                 float* __restrict__ Afused,
                                 float* __restrict__ biasf) {
    int o = threadIdx.x;
    if (o >= CDIM) return;
    float s2  = bn2_gamma[o] * rsqrtf(bn2_var[o] + EPS);
    float sh2 = bn2_beta[o] - bn2_mean[o] * s2;
    float acc = 0.f;
    for (int c = 0; c < CDIM; ++c) {
        float s1  = bn1_gamma[c] * rsqrtf(bn1_var[c] + EPS);
        float sh1 = bn1_beta[c] - bn1_mean[c] * s1;
        float w   = conv_w[o * CDIM + c];
        Afused[o * CDIM + c] = s2 * w * s1;
        acc += w * sh1;
    }
    biasf[o] = s2 * acc + sh2;
}

// ---------------------------------------------------------------------------
// Main kernel: one block per (p-chunk, b). 8 waves/block, each wave does
// TILES_PER_WAVE p-tiles of 16. Per tile: gather-max agg from LDS-staged
// feats, then 2 o-tiles x 8 k-steps of v_wmma_f32_16x16x4_f32, bias+relu,
// store.
// ---------------------------------------------------------------------------
__launch_bounds__(256)
__global__ void feyn_main(const float* __restrict__ feat_a,
                          const float* __restrict__ feat_b,
                          const int*   __restrict__ assign_a,
                          const int*   __restrict__ assign_b,
                          const float* __restrict__ Afused,
                          const float* __restrict__ biasf,
                          float* __restrict__ out) {
    __shared__ float lFA[CDIM * NA];    // 384 floats
    __shared__ float lFB[CDIM * NB];    // 384 floats
    __shared__ float lW[CDIM * CDIM];   // 1024 floats (fused weight, row-major [o][c])
    __shared__ float lBias[CDIM];

    const int b   = blockIdx.y;
    const int tid = threadIdx.x;

    // Stage per-batch feature tiles + fused params into LDS.
    const float* fa = feat_a + (size_t)b * CDIM * NA;
    const float* fb = feat_b + (size_t)b * CDIM * NB;
    for (int i = tid; i < CDIM * NA; i += 256) lFA[i] = fa[i];
    for (int i = tid; i < CDIM * NB; i += 256) lFB[i] = fb[i];
    for (int i = tid; i < CDIM * CDIM; i += 256) lW[i] = Afused[i];
    if (tid < CDIM) lBias[tid] = biasf[tid];
    __syncthreads();

    const int wave = tid >> 5;
    const int lane = tid & 31;
    const int half = lane >> 4;   // 0: K/M-rows low half, 1: high half
    const int n    = lane & 15;   // output column index within tile / A row

    for (int it = 0; it < TILES_PER_WAVE; ++it) {
        const int tile = blockIdx.x * (WAVES * TILES_PER_WAVE) + it * WAVES + wave;
        const int p0   = tile * 16;
        const int p    = p0 + n;

        const int ia0 = assign_a[p * 2 + 0];
        const int ia1 = assign_a[p * 2 + 1];
        const int ib0 = assign_b[p * 2 + 0];
        const int ib1 = assign_b[p * 2 + 1];

        // Build the 8 B-fragments (4x16 f32 each). For k-step s, this lane
        // holds rows K = {2*half, 2*half+1} -> channels c0 = 4s+2*half, c0+1.
        v2f bfrag[8];
#pragma unroll
        for (int s = 0; s < 8; ++s) {
            const int c0 = 4 * s + 2 * half;
            const int c1 = c0 + 1;
            float ga0 = fmaxf(lFA[c0 * NA + ia0], lFA[c0 * NA + ia1]);
            float gb0 = fmaxf(lFB[c0 * NB + ib0], lFB[c0 * NB + ib1]);
            float ga1 = fmaxf(lFA[c1 * NA + ia0], lFA[c1 * NA + ia1]);
            float gb1 = fmaxf(lFB[c1 * NB + ib0], lFB[c1 * NB + ib1]);
            bfrag[s].x = fmaxf(ga0, gb0);
            bfrag[s].y = fmaxf(ga1, gb1);
        }

        // Two 16x16 output o-tiles covering o = 0..31.
#pragma unroll
        for (int t = 0; t < 2; ++t) {
            const int o_row = t * 16 + n;     // A-matrix row M = lane&15
            v8f acc = {};
#pragma unroll
            for (int s = 0; s < 8; ++s) {
                v2f afrag;
                afrag.x = lW[o_row * CDIM + 4 * s + 2 * half];      // K = 2*half
                afrag.y = lW[o_row * CDIM + 4 * s + 1 + 2 * half];  // K = 2*half+1
                acc = __builtin_amdgcn_wmma_f32_16x16x4_f32(
                    /*neg_a=*/false, afrag, /*neg_b=*/false, bfrag[s],
                    /*c_mod=*/(short)0, acc, /*reuse_a=*/false, /*reuse_b=*/false);
            }
            // D VGPR v holds row o = t*16 + v + 8*half, column p = p0 + n.
#pragma unroll
            for (int v = 0; v < 8; ++v) {
                const int o = t * 16 + v + 8 * half;
                float y = acc[v] + lBias[o];
                y = fmaxf(y, 0.f);
                out[((size_t)b * CDIM + o) * PDIM + p] = y;
            }
        }
    }
}

extern "C" void kernel_launch(void* const* d_in, const int* in_sizes, int n_in,
                              void* d_out, int out_size, void* d_ws, size_t ws_size,
                              hipStream_t stream) {
    const float* feat_a    = (const float*)d_in[0];
    const float* feat_b    = (const float*)d_in[1];
    const int*   assign_a  = (const int*)d_in[2];
    const int*   assign_b  = (const int*)d_in[3];
    const float* bn1_gamma = (const float*)d_in[4];
    const float* bn1_beta  = (const float*)d_in[5];
    const float* bn1_mean  = (const float*)d_in[6];
    const float* bn1_var   = (const float*)d_in[7];
    const float* conv_w    = (const float*)d_in[8];
    const float* bn2_gamma = (const float*)d_in[9];
    const float* bn2_beta  = (const float*)d_in[10];
    const float* bn2_mean  = (const float*)d_in[11];
    const float* bn2_var   = (const float*)d_in[12];

    float* Afused = (float*)d_ws;              // 32*32 floats
    float* biasf  = Afused + CDIM * CDIM;      // 32 floats

    feyn_fuse_params<<<1, 32, 0, stream>>>(bn1_gamma, bn1_beta, bn1_mean, bn1_var,
                                           conv_w, bn2_gamma, bn2_beta, bn2_mean,
                                           bn2_var, Afused, biasf);

    dim3 grid(PDIM / (16 * WAVES * TILES_PER_WAVE), BDIM);  // (16, 128)
    feyn_main<<<grid, 256, 0, stream>>>(feat_a, feat_b, assign_a, assign_b,
                                        Afused, biasf, (float*)d_out);
}